// LambdaModel_52347061403800
// MI455X (gfx1250) — compile-verified
//
#include <hip/hip_runtime.h>

#define HID     256
#define DI      512
#define DSTATE  16
#define DCONV   4
#define DTRANK  16
#define LEN     2048
#define BATCH   4
#define NTOK    (BATCH * LEN)     /* 8192 tokens */
#define NCHUNK  16
#define CHUNKL  (LEN / NCHUNK)    /* 128 */
#define NEG_SLOPE 0.01f

typedef float v2f __attribute__((ext_vector_type(2)));
typedef float v8f __attribute__((ext_vector_type(8)));

// Fast activations: single v_exp_f32 / v_rcp_f32 instead of IEEE sequences.
__device__ __forceinline__ float fexp(float x)  { return __expf(x); }
__device__ __forceinline__ float sigm(float x)  { return __builtin_amdgcn_rcpf(1.0f + __expf(-x)); }
__device__ __forceinline__ float silu(float x)  { return x * sigm(x); }
__device__ __forceinline__ float softplus(float x)
{
    return (x > 20.0f) ? x : __logf(1.0f + __expf(x));
}

// ---------------------------------------------------------------------------
// Preprocess + proj0:  H[n,o] = b[o] + sum_i scale(inp[n,i]) * W[o,i]
// ---------------------------------------------------------------------------
__global__ void k_proj0(const float* __restrict__ inp, const float* __restrict__ w,
                        const float* __restrict__ b, float* __restrict__ H)
{
    int idx = blockIdx.x * blockDim.x + threadIdx.x;   // n*HID + o
    if (idx >= NTOK * HID) return;
    int o = idx & (HID - 1);
    int n = idx >> 8;
    float x0 = inp[n * 3 + 0] * (1.0f / 255.0f);
    float x1 = inp[n * 3 + 1] * (1.0f / (float)LEN);
    float x2 = inp[n * 3 + 2];
    H[idx] = b[o] + x0 * w[o * 3 + 0] + x1 * w[o * 3 + 1] + x2 * w[o * 3 + 2];
}

// Elementwise sigmoid (block-input activation, hoisted out of the GEMM K-loop)
__global__ void k_sigmoid(const float* __restrict__ in, float* __restrict__ out, int n)
{
    int idx = blockIdx.x * blockDim.x + threadIdx.x;
    if (idx < n) out[idx] = sigm(in[idx]);
}

// ---------------------------------------------------------------------------
// fp32 WMMA GEMM:  C[NTOK,N] = actOut( A[NTOK,K] @ W[N,K]^T + bias )
// All shapes/strides compile-time -> loads are [base + imm] with no per-iter
// address math. One wave computes a 16 x (16*NT) strip with NT accumulators;
// the K-loop is software-pipelined (next fragments prefetched into registers
// before the current NT WMMAs issue).  actOut: 0=none 1=softplus 2=leaky-relu
// ---------------------------------------------------------------------------
template <int NT, int N, int K, int LDA, int LDW, int LDC>
__global__ void k_gemm_wmma(const float* __restrict__ A,
                            const float* __restrict__ W,
                            const float* __restrict__ bias,
                            float* __restrict__ C, int actOut)
{
    constexpr int NSTRIPS = N / (16 * NT);
    int gtid = blockIdx.x * blockDim.x + threadIdx.x;
    int wave = gtid >> 5;
    int lane = threadIdx.x & 31;
    int mt = wave / NSTRIPS;
    int ns = wave - mt * NSTRIPS;
    if (mt * 16 >= NTOK) return;
    int m0 = mt << 4, n0 = ns * 16 * NT;

    int l15  = lane & 15;
    int half = lane >> 4;            // 0: K pair {0,1}, 1: K pair {2,3}
    int kb   = half * 2;

    const float* arow = A + (size_t)(m0 + l15) * LDA + kb;
    const float* wr   = W + (size_t)(n0 + l15) * LDW + kb;

    v8f acc[NT];
    #pragma unroll
    for (int t = 0; t < NT; ++t) acc[t] = (v8f){};

    // prologue: fragments for k0 = 0
    v2f av, bv[NT];
    av.x = arow[0];
    av.y = arow[1];
    #pragma unroll
    for (int t = 0; t < NT; ++t) {
        bv[t].x = wr[t * 16 * LDW + 0];
        bv[t].y = wr[t * 16 * LDW + 1];
    }

    #pragma unroll 4
    for (int k0 = 0; k0 < K - 4; k0 += 4) {
        // prefetch next k-step's fragments (overlaps with WMMAs below)
        v2f avn, bvn[NT];
        avn.x = arow[k0 + 4];
        avn.y = arow[k0 + 5];
        #pragma unroll
        for (int t = 0; t < NT; ++t) {
            bvn[t].x = wr[t * 16 * LDW + k0 + 4];
            bvn[t].y = wr[t * 16 * LDW + k0 + 5];
        }
        #pragma unroll
        for (int t = 0; t < NT; ++t)
            acc[t] = __builtin_amdgcn_wmma_f32_16x16x4_f32(
                         false, av, false, bv[t], (short)0, acc[t], false, false);
        av = avn;
        #pragma unroll
        for (int t = 0; t < NT; ++t) bv[t] = bvn[t];
    }
    #pragma unroll
    for (int t = 0; t < NT; ++t)
        acc[t] = __builtin_amdgcn_wmma_f32_16x16x4_f32(
                     false, av, false, bv[t], (short)0, acc[t], false, false);

    // C/D layout: VGPR v -> rows m0+v (lanes 0-15) and m0+v+8 (lanes 16-31)
    #pragma unroll
    for (int t = 0; t < NT; ++t) {
        int cn = n0 + t * 16 + l15;
        float bvv = bias ? bias[cn] : 0.0f;
        #pragma unroll
        for (int v = 0; v < 8; ++v) {
            int r = m0 + v + half * 8;
            float val = acc[t][v] + bvv;
            if (actOut == 1)      val = softplus(val);
            else if (actOut == 2) val = (val >= 0.0f) ? val : NEG_SLOPE * val;
            C[(size_t)r * LDC + cn] = val;
        }
    }
}

// ---------------------------------------------------------------------------
// Depthwise causal conv (DCONV=4, left pad 3) + bias + SiLU.
// xin = XZ[:, 0:DI] (row stride 2*DI) -> XS[NTOK, DI]
// ---------------------------------------------------------------------------
__global__ void k_conv_silu(const float* __restrict__ XZ, const float* __restrict__ cw,
                            const float* __restrict__ cb, float* __restrict__ XS)
{
    int idx = blockIdx.x * blockDim.x + threadIdx.x;  // n*DI + c
    if (idx >= NTOK * DI) return;
    int c = idx & (DI - 1);
    int n = idx >> 9;
    int t = n & (LEN - 1);
    float acc = cb[c];
    #pragma unroll
    for (int j = 0; j < DCONV; ++j) {
        int tt = t + j - (DCONV - 1);
        if (tt >= 0)
            acc += XZ[(size_t)(n + j - (DCONV - 1)) * (2 * DI) + c] * cw[c * DCONV + j];
    }
    XS[idx] = silu(acc);
}

// ---------------------------------------------------------------------------
// Selective scan, 3-phase chunked linear recurrence.
// tid layout (phases A/C): (((b*DI + d)*NCHUNK + c)*DSTATE) + s
// ---------------------------------------------------------------------------
__global__ void k_scan_A(const float* __restrict__ DT, const float* __restrict__ XDBL,
                         const float* __restrict__ XS, const float* __restrict__ A_log,
                         float* __restrict__ APROD, float* __restrict__ BACC)
{
    int tid = blockIdx.x * blockDim.x + threadIdx.x;
    if (tid >= BATCH * DI * NCHUNK * DSTATE) return;
    int s = tid & 15;
    int c = (tid >> 4) & (NCHUNK - 1);
    int d = (tid >> 8) & (DI - 1);
    int b = tid >> 17;
    float Ac = -fexp(A_log[d * DSTATE + s]);
    float ap = 1.0f, h = 0.0f;
    int tbase = c * CHUNKL;
    for (int i = 0; i < CHUNKL; ++i) {
        int n = b * LEN + tbase + i;
        float dt = DT[(size_t)n * DI + d];
        float a  = fexp(dt * Ac);
        float bm = XDBL[(size_t)n * 48 + DTRANK + s];
        float bx = dt * bm * XS[(size_t)n * DI + d];
        h  = a * h + bx;
        ap *= a;
    }
    APROD[tid] = ap;
    BACC[tid]  = h;
}

__global__ void k_scan_B(const float* __restrict__ APROD, const float* __restrict__ BACC,
                         float* __restrict__ HST)
{
    int tid = blockIdx.x * blockDim.x + threadIdx.x;   // (b*DI + d)*16 + s
    if (tid >= BATCH * DI * DSTATE) return;
    int s  = tid & 15;
    int bd = tid >> 4;
    float h = 0.0f;
    for (int c = 0; c < NCHUNK; ++c) {
        int idx = (bd * NCHUNK + c) * DSTATE + s;
        HST[idx] = h;                       // state BEFORE chunk c
        h = APROD[idx] * h + BACC[idx];
    }
}

// Phase C with fused gating: lane s==0 writes XS[n,d] = (y + D*xs) * silu(z)
__global__ void k_scan_C(const float* __restrict__ DT, const float* __restrict__ XDBL,
                         const float* __restrict__ A_log, const float* __restrict__ HST,
                         const float* __restrict__ Dp, const float* __restrict__ XZ,
                         float* __restrict__ XS)
{
    int tid = blockIdx.x * blockDim.x + threadIdx.x;
    if (tid >= BATCH * DI * NCHUNK * DSTATE) return;
    int s = tid & 15;
    int c = (tid >> 4) & (NCHUNK - 1);
    int d = (tid >> 8) & (DI - 1);
    int b = tid >> 17;
    float Ac = -fexp(A_log[d * DSTATE + s]);
    float Dd = Dp[d];
    float h  = HST[tid];
    int tbase = c * CHUNKL;
    for (int i = 0; i < CHUNKL; ++i) {
        int n = b * LEN + tbase + i;
        float dt  = DT[(size_t)n * DI + d];
        float a   = fexp(dt * Ac);
        float bm  = XDBL[(size_t)n * 48 + DTRANK + s];
        float xsv = XS[(size_t)n * DI + d];
        h = a * h + dt * bm * xsv;
        float yv = h * XDBL[(size_t)n * 48 + DTRANK + DSTATE + s];
        // reduce over the 16 DSTATE lanes (stays within 16-lane halves of wave32)
        yv += __shfl_xor(yv, 1);
        yv += __shfl_xor(yv, 2);
        yv += __shfl_xor(yv, 4);
        yv += __shfl_xor(yv, 8);
        if (s == 0) {
            float z = XZ[(size_t)n * (2 * DI) + DI + d];
            XS[(size_t)n * DI + d] = (yv + Dd * xsv) * silu(z);
        }
    }
}

// ---------------------------------------------------------------------------
// Head: one wave per batch element; shfl-butterfly reductions.
// ---------------------------------------------------------------------------
__device__ __forceinline__ float wave_sum(float v)
{
    v += __shfl_xor(v, 1);
    v += __shfl_xor(v, 2);
    v += __shfl_xor(v, 4);
    v += __shfl_xor(v, 8);
    v += __shfl_xor(v, 16);
    return v;
}

__device__ __forceinline__ float stepfn_wave(float x, float lowest, int steps,
                                             float width, int lane)
{
    float base = x + steps * width * 0.5f;
    float part = 0.0f;
    for (int n = 1 + lane; n <= steps; n += 32)
        part += sigm(base - width * (float)n);
    return lowest + wave_sum(part);
}

__device__ __forceinline__ float nan0(float v) { return isfinite(v) ? v : 0.0f; }

__global__ void k_head(const float* __restrict__ H, const float* __restrict__ inp,
                       const float* __restrict__ p1w, const float* __restrict__ p1b,
                       float* __restrict__ out)
{
    int b    = threadIdx.x >> 5;    // block of 128 = 4 waves, one per batch
    int lane = threadIdx.x & 31;
    if (b >= BATCH) return;
    const float* hrow = H + (size_t)(b * LEN + LEN - 1) * HID;

    float o[5];
    #pragma unroll
    for (int j = 0; j < 5; ++j) {
        float part = 0.0f;
        for (int i = lane; i < HID; i += 32)
            part += hrow[i] * p1w[j * HID + i];
        o[j] = p1b[j] + wave_sum(part);   // every lane holds the full sum
    }

    float s0 = stepfn_wave(o[0], -71.0f, 144, 24.0f, lane);
    float s1 = stepfn_wave(o[1], -63.0f, 128, 24.0f, lane);
    float s2 = stepfn_wave(o[2], -63.0f, 128, 24.0f, lane);
    float s3 = stepfn_wave(o[3], -63.0f, 128, 24.0f, lane);
    float s4 = stepfn_wave(o[4], -63.0f, 128, 24.0f, lane);

    if (lane == 0) {
        float inp_last0  = inp[(size_t)(b * LEN + LEN - 1) * 3 + 0];
        float inp_last1  = inp[(size_t)(b * LEN + LEN - 1) * 3 + 1];
        float inp_first2 = inp[(size_t)(b * LEN) * 3 + 2];
        out[b * 3 + 0] = inp_last0 + s0;
        out[b * 3 + 1] = inp_last1 + (inp_first2 + 0.001f) * nan0(s1 / s2);
        out[b * 3 + 2] = inp_first2 * nan0(s3 / s4);
    }
}

// ---------------------------------------------------------------------------
// Host launcher
// ---------------------------------------------------------------------------
extern "C" void kernel_launch(void* const* d_in, const int* in_sizes, int n_in,
                              void* d_out, int out_size, void* d_ws, size_t ws_size,
                              hipStream_t stream)
{
    (void)in_sizes; (void)n_in; (void)out_size; (void)ws_size;

    const float* inp = (const float*)d_in[0];
    const float* p0w = (const float*)d_in[1];
    const float* p0b = (const float*)d_in[2];
    const float* p1w = (const float*)d_in[25];
    const float* p1b = (const float*)d_in[26];

    float* ws   = (float*)d_ws;
    float* XZ   = ws;                         // NTOK*1024
    float* XS   = XZ   + (size_t)NTOK * 1024; // NTOK*512
    float* DT   = XS   + (size_t)NTOK * DI;   // NTOK*512
    float* XDBL = DT   + (size_t)NTOK * DI;   // NTOK*48
    float* H    = XDBL + (size_t)NTOK * 48;   // NTOK*256
    float* TH   = H    + (size_t)NTOK * HID;  // NTOK*256 (also sigmoid(H) buffer)
    float* APROD= TH   + (size_t)NTOK * HID;  // 524288
    float* BACC = APROD+ (size_t)BATCH * DI * NCHUNK * DSTATE;
    float* HST  = BACC + (size_t)BATCH * DI * NCHUNK * DSTATE;

    const int TPB = 256;

    k_proj0<<<NTOK * HID / TPB, TPB, 0, stream>>>(inp, p0w, p0b, H);

    for (int blk = 0; blk < 2; ++blk) {
        int base = 3 + blk * 11;
        const float* w_inproj = (const float*)d_in[base + 0];
        const float* conv_w   = (const float*)d_in[base + 1];
        const float* conv_b   = (const float*)d_in[base + 2];
        const float* w_xproj  = (const float*)d_in[base + 3];
        const float* dt_w     = (const float*)d_in[base + 4];
        const float* dt_b     = (const float*)d_in[base + 5];
        const float* A_log    = (const float*)d_in[base + 6];
        const float* Dp       = (const float*)d_in[base + 7];
        const float* w_outp   = (const float*)d_in[base + 8];
        const float* lin_w    = (const float*)d_in[base + 9];
        const float* lin_b    = (const float*)d_in[base + 10];

        // block input activation: TH = sigmoid(H)
        k_sigmoid<<<NTOK * HID / TPB, TPB, 0, stream>>>(H, TH, NTOK * HID);

        // in_proj: XZ = TH @ W^T   (8192x256 -> 8192x1024)
        {
            int waves = (NTOK / 16) * (1024 / 64);           // 8192
            k_gemm_wmma<4, 1024, 256, 256, 256, 1024>
                <<<waves * 32 / TPB, TPB, 0, stream>>>(TH, w_inproj, nullptr, XZ, 0);
        }
        // conv + silu -> XS
        k_conv_silu<<<NTOK * DI / TPB, TPB, 0, stream>>>(XZ, conv_w, conv_b, XS);
        // x_proj: XDBL = XS @ W^T  (8192x512 -> 8192x48), NT=3 (full row strip)
        {
            int waves = (NTOK / 16);                         // 512
            k_gemm_wmma<3, 48, 512, 512, 512, 48>
                <<<waves * 32 / TPB, TPB, 0, stream>>>(XS, w_xproj, nullptr, XDBL, 0);
        }
        // dt: DT = softplus(XDBL[:, :16] @ dt_w^T + dt_b)  (8192x16 -> 8192x512)
        {
            int waves = (NTOK / 16) * (512 / 64);            // 4096
            k_gemm_wmma<4, 512, 16, 48, 16, 512>
                <<<waves * 32 / TPB, TPB, 0, stream>>>(XDBL, dt_w, dt_b, DT, 1);
        }
        // selective scan (chunked); phase C fuses the gating into XS
        k_scan_A<<<BATCH * DI * NCHUNK * DSTATE / TPB, TPB, 0, stream>>>(
            DT, XDBL, XS, A_log, APROD, BACC);
        k_scan_B<<<BATCH * DI * DSTATE / TPB, TPB, 0, stream>>>(APROD, BACC, HST);
        k_scan_C<<<BATCH * DI * NCHUNK * DSTATE / TPB, TPB, 0, stream>>>(
            DT, XDBL, A_log, HST, Dp, XZ, XS);
        // out_proj + leaky relu: TH = lrelu(XS @ W^T)  (8192x512 -> 8192x256)
        {
            int waves = (NTOK / 16) * (256 / 64);            // 2048
            k_gemm_wmma<4, 256, 512, 512, 512, 256>
                <<<waves * 32 / TPB, TPB, 0, stream>>>(XS, w_outp, nullptr, TH, 2);
        }
        // lin: H = TH @ lin_w^T + lin_b  (8192x256 -> 8192x256)
        {
            int waves = (NTOK / 16) * (256 / 64);            // 2048
            k_gemm_wmma<4, 256, 256, 256, 256, 256>
                <<<waves * 32 / TPB, TPB, 0, stream>>>(TH, lin_w, lin_b, H, 0);
        }
    }

    k_head<<<1, 128, 0, stream>>>(H, inp, p1w, p1b, (float*)d_out);
}